// ExemplarSoftmaxLoss_17102559773292
// MI455X (gfx1250) — compile-verified
//
#include <hip/hip_runtime.h>
#include <cstdint>
#include <cstddef>

// ---------------------------------------------------------------------------
// ExemplarSoftmaxLoss for MI455X (gfx1250, wave32).
// Memory-bound streaming reduction: ~300 MB traffic -> ~13 us at 23.3 TB/s.
// No GEMM structure -> no WMMA; CDNA5 features used instead:
//   * async global->LDS staging (ASYNCcnt) for the softmax rows
//   * wave32 shuffle reductions
//   * global_prefetch_b8 for the gathered exemplar rows
//   * non-temporal (th:NT) loads for read-once streams to keep the exemplar
//     table resident in the 192 MB L2
// ---------------------------------------------------------------------------

#define MARGIN_  1.0f
#define MARGIN2_ 0.2f
#define LAMBDA_  1.0f
#define EPS_     1e-6f

constexpr int kC   = 1000;     // classes
constexpr int kC4  = kC / 4;   // 250 float4 per outputs row
constexpr int kD   = 512;      // feature dim
constexpr int kWavesPerBlock = 8;

// Pointer/vector types matching the async-LDS builtin signature
// (v4i as1*, v4i as3*, imm offset, imm cpol) confirmed in round 1/2.
typedef int   v4i __attribute__((vector_size(16)));
typedef float vf4 __attribute__((vector_size(16)));
typedef __attribute__((address_space(1))) v4i* gv4i_t;
typedef __attribute__((address_space(3))) v4i* lv4i_t;

#if defined(__gfx1250__) && __has_builtin(__builtin_amdgcn_global_load_async_to_lds_b128)
#define USE_ASYNC_LDS 1
#else
#define USE_ASYNC_LDS 0
#endif

__device__ __forceinline__ void wait_async0() {
#if defined(__gfx1250__)
#if __has_builtin(__builtin_amdgcn_s_wait_asynccnt)
  __builtin_amdgcn_s_wait_asynccnt(0);
  asm volatile("" ::: "memory");
#else
  asm volatile("s_wait_asynccnt 0" ::: "memory");
#endif
#endif
}

__device__ __forceinline__ float wave_max_f32(float v) {
#pragma unroll
  for (int off = 16; off > 0; off >>= 1) v = fmaxf(v, __shfl_xor(v, off, 32));
  return v;
}

__device__ __forceinline__ float wave_sum_f32(float v) {
#pragma unroll
  for (int off = 16; off > 0; off >>= 1) v += __shfl_xor(v, off, 32);
  return v;
}

// ---------------------------------------------------------------------------
// Kernel A: per-row logsumexp of MARGIN*outputs; accumulate -logp[label].
// One wave per row; row (4000 B) async-staged into LDS via B128 async loads.
// ---------------------------------------------------------------------------
__global__ __launch_bounds__(256)
void softmax_rows_kernel(const float* __restrict__ outputs,
                         const int* __restrict__ labels_a,
                         const int* __restrict__ labels_n,
                         float* __restrict__ ws, int B, int rows) {
  __shared__ float4 rowbuf[kWavesPerBlock][256];  // 250 used, padded to 256
  __shared__ float bsum;

  const int wave = threadIdx.x >> 5;
  const int lane = threadIdx.x & 31;
  if (threadIdx.x == 0) bsum = 0.0f;
  __syncthreads();

  const int row = blockIdx.x * kWavesPerBlock + wave;
  if (row < rows) {
    const float4* __restrict__ src = (const float4*)(outputs + (size_t)row * kC);

#if USE_ASYNC_LDS
#pragma unroll
    for (int k = 0; k < 8; ++k) {
      const int c = lane + (k << 5);
      if (c < kC4) {
        __builtin_amdgcn_global_load_async_to_lds_b128(
            (gv4i_t)(src + c), (lv4i_t)&rowbuf[wave][c], 0, 0);
      }
    }
    wait_async0();
#else
#pragma unroll
    for (int k = 0; k < 8; ++k) {
      const int c = lane + (k << 5);
      if (c < kC4) rowbuf[wave][c] = src[c];
    }
#endif

    // Pass 1: row max of MARGIN*x
    float mx = -__builtin_huge_valf();
#pragma unroll
    for (int k = 0; k < 8; ++k) {
      const int c = lane + (k << 5);
      if (c < kC4) {
        const float4 v = rowbuf[wave][c];
        mx = fmaxf(mx, fmaxf(fmaxf(MARGIN_ * v.x, MARGIN_ * v.y),
                             fmaxf(MARGIN_ * v.z, MARGIN_ * v.w)));
      }
    }
    mx = wave_max_f32(mx);

    // Pass 2: sum exp(MARGIN*x - mx), native v_exp_f32 path
    float s = 0.0f;
#pragma unroll
    for (int k = 0; k < 8; ++k) {
      const int c = lane + (k << 5);
      if (c < kC4) {
        const float4 v = rowbuf[wave][c];
        s += __expf(fmaf(MARGIN_, v.x, -mx));
        s += __expf(fmaf(MARGIN_, v.y, -mx));
        s += __expf(fmaf(MARGIN_, v.z, -mx));
        s += __expf(fmaf(MARGIN_, v.w, -mx));
      }
    }
    s = wave_sum_f32(s);

    if (lane == 0) {
      const int label = (row < B) ? labels_a[row]
                      : (row < 2 * B) ? labels_a[row - B]
                                      : labels_n[row - 2 * B];
      const float* rf = (const float*)&rowbuf[wave][0];
      const float xl = MARGIN_ * rf[label];
      // -log_softmax at label = -(xl - mx - log(sum))
      atomicAdd(&bsum, -(xl - mx - __logf(s)));
    }
  }
  __syncthreads();
  if (threadIdx.x == 0) atomicAdd(&ws[0], bsum);
}

// ---------------------------------------------------------------------------
// Kernel B: six fused L2 distances per row + center/triplet hinge terms.
// One wave per row; each lane owns 4 float4 chunks of each vector.
// anchor/positive/negative are read-once -> non-temporal loads; exemplar
// gathers stay regular-temporal so the 2 MB table lives in L2.
// ---------------------------------------------------------------------------
__global__ __launch_bounds__(256)
void dist_loss_kernel(const float* __restrict__ anchor,
                      const float* __restrict__ positive,
                      const float* __restrict__ negative,
                      const int* __restrict__ labels_a,
                      const int* __restrict__ labels_n,
                      const float* __restrict__ exemplars,
                      float* __restrict__ ws, int B) {
  __shared__ float bcent, btrip;
  if (threadIdx.x == 0) { bcent = 0.0f; btrip = 0.0f; }
  __syncthreads();

  const int wave = threadIdx.x >> 5;
  const int lane = threadIdx.x & 31;
  const int row = blockIdx.x * kWavesPerBlock + wave;

  if (row < B) {
    const int ia = labels_a[row];
    const int ib = labels_n[row];
    const vf4* __restrict__ a4  = (const vf4*)(anchor    + (size_t)row * kD);
    const vf4* __restrict__ p4  = (const vf4*)(positive  + (size_t)row * kD);
    const vf4* __restrict__ n4  = (const vf4*)(negative  + (size_t)row * kD);
    const vf4* __restrict__ ea4 = (const vf4*)(exemplars + (size_t)ia  * kD);
    const vf4* __restrict__ en4 = (const vf4*)(exemplars + (size_t)ib  * kD);

    // Pre-touch the gathered exemplar rows (2 KB each, one line per lane)
    // -> global_prefetch_b8 on gfx1250.
    __builtin_prefetch(((const char*)ea4) + lane * 64, 0, 3);
    __builtin_prefetch(((const char*)en4) + lane * 64, 0, 3);

    float s1 = 0.f, s2 = 0.f, s3 = 0.f, s4 = 0.f, s5 = 0.f, s6 = 0.f;
#pragma unroll
    for (int k = 0; k < 4; ++k) {
      const int c = lane + (k << 5);
      const vf4 av = __builtin_nontemporal_load(a4 + c);
      const vf4 pv = __builtin_nontemporal_load(p4 + c);
      const vf4 nv = __builtin_nontemporal_load(n4 + c);
      const vf4 eav = ea4[c];
      const vf4 env = en4[c];
#define ACC_COMP(i)                                                        \
      {                                                                    \
        const float da = av[i] - eav[i] + EPS_; s1 = fmaf(da, da, s1);     \
        const float dn = nv[i] - eav[i] + EPS_; s2 = fmaf(dn, dn, s2);     \
        const float db = av[i] - env[i] + EPS_; s3 = fmaf(db, db, s3);     \
        const float dm = nv[i] - env[i] + EPS_; s4 = fmaf(dm, dm, s4);     \
        const float dp = av[i] - pv[i]  + EPS_; s5 = fmaf(dp, dp, s5);     \
        const float dq = av[i] - nv[i]  + EPS_; s6 = fmaf(dq, dq, s6);     \
      }
      ACC_COMP(0) ACC_COMP(1) ACC_COMP(2) ACC_COMP(3)
#undef ACC_COMP
    }

    s1 = wave_sum_f32(s1);
    s2 = wave_sum_f32(s2);
    s3 = wave_sum_f32(s3);
    s4 = wave_sum_f32(s4);
    s5 = wave_sum_f32(s5);
    s6 = wave_sum_f32(s6);

    if (lane == 0) {
      const float d_ref1 = sqrtf(s1);   // ||a - ex_a||
      const float d_neg1 = sqrtf(s2);   // ||n - ex_a||
      const float d_ref2 = sqrtf(s3);   // ||a - ex_n||
      const float d_neg2 = sqrtf(s4);   // ||n - ex_n||
      const float tp     = sqrtf(s5);   // ||a - p||
      const float tn     = sqrtf(s6);   // ||a - n||

      const float f1 = d_ref1 - d_neg1;
      const float c1 = (f1 > 0.0f) ? (f1 + MARGIN2_) : 0.0f;
      const float f2 = d_neg2 - d_ref2;
      const float c2 = (f2 > 0.0f) ? f2 : 0.0f;
      const float ft = tp - tn;
      const float t  = (ft > 0.0f) ? ft : 0.0f;

      atomicAdd(&bcent, c1 + c2);
      atomicAdd(&btrip, t);
    }
  }
  __syncthreads();
  if (threadIdx.x == 0) {
    atomicAdd(&ws[1], bcent);
    atomicAdd(&ws[2], btrip);
  }
}

// ---------------------------------------------------------------------------
// Kernel C: combine scalars.  d_out = (total, triplet, softmax, center)
// ---------------------------------------------------------------------------
__global__ void finalize_kernel(const float* __restrict__ ws,
                                float* __restrict__ out, int B) {
  const float sm = ws[0] / (3.0f * (float)B);
  const float ce = ws[1];
  const float tr = ws[2];
  out[0] = sm + 0.01f * ce + LAMBDA_ * tr;
  out[1] = tr;
  out[2] = sm;
  out[3] = ce;
}

extern "C" void kernel_launch(void* const* d_in, const int* in_sizes, int n_in,
                              void* d_out, int out_size, void* d_ws, size_t ws_size,
                              hipStream_t stream) {
  (void)n_in; (void)out_size; (void)ws_size;
  const float* anchor   = (const float*)d_in[0];
  const float* positive = (const float*)d_in[1];
  const float* negative = (const float*)d_in[2];
  const float* outputs  = (const float*)d_in[3];
  const int*   la       = (const int*)d_in[4];
  const int*   ln       = (const int*)d_in[5];
  const float* ex       = (const float*)d_in[6];
  float* out = (float*)d_out;
  float* ws  = (float*)d_ws;

  const int B    = in_sizes[4];   // 16384
  const int rows = 3 * B;         // 49152

  (void)hipMemsetAsync(ws, 0, 3 * sizeof(float), stream);

  const dim3 blk(256);
  softmax_rows_kernel<<<dim3((rows + kWavesPerBlock - 1) / kWavesPerBlock), blk, 0,
                        stream>>>(outputs, la, ln, ws, B, rows);
  dist_loss_kernel<<<dim3((B + kWavesPerBlock - 1) / kWavesPerBlock), blk, 0,
                     stream>>>(anchor, positive, negative, la, ln, ex, ws, B);
  finalize_kernel<<<1, 1, 0, stream>>>(ws, out, B);
}